// GATNetv1_41721312313875
// MI455X (gfx1250) — compile-verified
//
#include <hip/hip_runtime.h>
#include <hip/hip_bf16.h>

// ---------------------------------------------------------------------------
// GAT network forward for MI455X (gfx1250, wave32).
// All bn->matmul pairs folded into fp32 WMMA GEMMs (v_wmma_f32_16x16x4_f32).
// ---------------------------------------------------------------------------

typedef __attribute__((ext_vector_type(2))) float v2f;
typedef __attribute__((ext_vector_type(8))) float v8f;

#define HID 128
#define HEADS 4
#define DHEAD 32
#define NCLS 10
#define NGRAPHS 256
#define LDS_STRIDE 130   // padded stride kills bank conflicts on transposed stores

// ------------------------- generic fill ------------------------------------
__global__ void fill_u32_kernel(unsigned* __restrict__ p, unsigned v, int n) {
    int t = blockIdx.x * blockDim.x + threadIdx.x;
    if (t < n) p[t] = v;
}

// ------------------------- BN column stats ---------------------------------
// blockDim = 128 (one thread per feature column), 128 rows per block.
__global__ void colsum_kernel(const float* __restrict__ x, int N,
                              float* __restrict__ sum, float* __restrict__ sumsq) {
    int c  = threadIdx.x;
    int r0 = blockIdx.x * 128;
    int r1 = r0 + 128; if (r1 > N) r1 = N;
    float s = 0.f, q = 0.f;
    for (int r = r0; r < r1; ++r) {
        float v = x[r * HID + c];
        s += v; q += v * v;
    }
    atomicAdd(&sum[c], s);
    atomicAdd(&sumsq[c], q);
}

// scale = g * rsqrt(var + eps); shift = b - mean * scale   (biased var, ddof=0)
__global__ void bn_scale_shift_kernel(const float* __restrict__ sum,
                                      const float* __restrict__ sumsq,
                                      const float* __restrict__ gamma,
                                      const float* __restrict__ beta,
                                      float invN,
                                      float* __restrict__ scale,
                                      float* __restrict__ shift) {
    int c = threadIdx.x;
    float m  = sum[c] * invN;
    float v  = sumsq[c] * invN - m * m;
    float sc = gamma[c] * rsqrtf(v + 1e-5f);
    scale[c] = sc;
    shift[c] = beta[c] - m * sc;
}

// Fold BN affine into weight: Wout[k][j] = scale[k]*W[k][j];
// bias_out[j] = (bias_in?bias_in[j]:0) + sum_k shift[k]*W[k][j]
__global__ void fold_weight_kernel(const float* __restrict__ W,
                                   const float* __restrict__ scale,
                                   const float* __restrict__ shift,
                                   const float* __restrict__ bias_in,
                                   float* __restrict__ Wout,
                                   float* __restrict__ bias_out) {
    int j = threadIdx.x;                 // 128 threads, 1 block
    float acc = bias_in ? bias_in[j] : 0.f;
    for (int k = 0; k < HID; ++k) {
        float w = W[k * HID + j];
        Wout[k * HID + j] = scale[k] * w;
        acc += shift[k] * w;
    }
    bias_out[j] = acc;
}

// ------------------------- WMMA fp32 GEMM ----------------------------------
// Y[M,128] = relu?( X[M,128] @ W[128,128] + bias )
// 4 waves / block; wave w computes rows [blk*64 + w*16, +16) x all 128 cols.
// Per-wave: 8 accumulators of v8f, K stepped by 4 via v_wmma_f32_16x16x4_f32.
__global__ __launch_bounds__(128) void gemm128_wmma_kernel(
        const float* __restrict__ X, const float* __restrict__ W,
        const float* __restrict__ bias, float* __restrict__ Y,
        int M, int relu) {
    __shared__ float Wt[HID * LDS_STRIDE];   // transposed: Wt[col][k]
    const int tid = threadIdx.x;

    // cooperative transposed stage of W into LDS (thread tid owns column tid)
    for (int i = tid; i < HID * HID; i += 128) {
        int k = i >> 7;          // row of W
        int c = i & 127;         // col of W (== tid)
        Wt[c * LDS_STRIDE + k] = W[i];
    }
    __syncthreads();

    const int wave = tid >> 5;
    const int lane = tid & 31;
    const int row0 = blockIdx.x * 64 + wave * 16;
    if (row0 >= M) return;                       // wave-uniform exit (EXEC stays full)

    const int half = lane >> 4;                  // 0: K={k,k+1}; 1: K={k+2,k+3}
    const int l16  = lane & 15;
    const int arow = row0 + l16;

    v8f acc[8] = {};                             // zero-init accumulators
    for (int k = 0; k < HID; k += 4) {
        const int kk = k + 2 * half;
        // A fragment (16x4): lane carries X[arow][kk], X[arow][kk+1]
        const v2f a = *(const v2f*)(X + (size_t)arow * HID + kk);
#pragma unroll
        for (int t = 0; t < 8; ++t) {
            const int col = t * 16 + l16;
            // B fragment (4x16): lane carries W[kk][col], W[kk+1][col]
            const v2f b = *(const v2f*)(&Wt[col * LDS_STRIDE + kk]);
            acc[t] = __builtin_amdgcn_wmma_f32_16x16x4_f32(
                false, a, false, b, (short)0, acc[t], false, false);
        }
    }

    // C/D layout: lane<16 -> rows row0+0..7 ; lane>=16 -> rows row0+8..15
    const int mbase = half * 8;
#pragma unroll
    for (int t = 0; t < 8; ++t) {
        const int col = t * 16 + l16;
        const float bv = bias ? bias[col] : 0.f;
#pragma unroll
        for (int r = 0; r < 8; ++r) {
            float v = acc[t][r] + bv;
            if (relu) v = fmaxf(v, 0.f);
            Y[(size_t)(row0 + mbase + r) * HID + col] = v;
        }
    }
}

// ------------------------- attention alphas --------------------------------
__global__ void alpha_kernel(const float* __restrict__ hp,
                             const float* __restrict__ asrc,
                             const float* __restrict__ adst,
                             float* __restrict__ als, float* __restrict__ ald,
                             int N) {
    int t = blockIdx.x * blockDim.x + threadIdx.x;     // t = n*HEADS + head
    if (t >= N * HEADS) return;
    int head = t & (HEADS - 1);
    int n    = t >> 2;
    const float* hv = hp + (size_t)n * HID + head * DHEAD;
    const float* av = asrc + head * DHEAD;
    const float* bv = adst + head * DHEAD;
    float s = 0.f, d = 0.f;
#pragma unroll 8
    for (int i = 0; i < DHEAD; ++i) {
        float h = hv[i];
        s += h * av[i];
        d += h * bv[i];
    }
    als[t] = s;
    ald[t] = d;
}

// ------------------------- segment softmax over edges ----------------------
__device__ __forceinline__ unsigned fkey(float f) {
    unsigned b = __float_as_uint(f);
    return (b & 0x80000000u) ? ~b : (b | 0x80000000u);
}
__device__ __forceinline__ float funkey(unsigned k) {
    unsigned b = (k & 0x80000000u) ? (k & 0x7FFFFFFFu) : ~k;
    return __uint_as_float(b);
}
__device__ __forceinline__ void edge_nodes(const int* ei, int nE, int e,
                                           int& s, int& d) {
    if (e < nE) { s = ei[e]; d = ei[nE + e]; }
    else        { s = d = e - nE; }            // appended self-loops
}
__device__ __forceinline__ float lrelu02(float x) {
    return x > 0.f ? x : 0.2f * x;
}

__global__ void edge_max_kernel(const int* __restrict__ ei, int nE, int N,
                                const float* __restrict__ als,
                                const float* __restrict__ ald,
                                unsigned* __restrict__ mkey) {
    int t = blockIdx.x * blockDim.x + threadIdx.x;     // t = e*HEADS + head
    if (t >= (nE + N) * HEADS) return;
    int head = t & (HEADS - 1), e = t >> 2;
    int s, d; edge_nodes(ei, nE, e, s, d);
    float l = lrelu02(als[s * HEADS + head] + ald[d * HEADS + head]);
    atomicMax(&mkey[d * HEADS + head], fkey(l));
}

__global__ void edge_exp_kernel(const int* __restrict__ ei, int nE, int N,
                                const float* __restrict__ als,
                                const float* __restrict__ ald,
                                const unsigned* __restrict__ mkey,
                                float* __restrict__ ebuf,
                                float* __restrict__ ssum) {
    int t = blockIdx.x * blockDim.x + threadIdx.x;
    if (t >= (nE + N) * HEADS) return;
    int head = t & (HEADS - 1), e = t >> 2;
    int s, d; edge_nodes(ei, nE, e, s, d);
    float l  = lrelu02(als[s * HEADS + head] + ald[d * HEADS + head]);
    float m  = funkey(mkey[d * HEADS + head]);
    float ev = expf(l - m);
    ebuf[t] = ev;
    atomicAdd(&ssum[d * HEADS + head], ev);
}

// one wave per edge; lane owns 4 consecutive feature floats
__global__ __launch_bounds__(256) void scatter_kernel(
        const int* __restrict__ ei, int nE, int N,
        const float* __restrict__ hp, const float* __restrict__ ebuf,
        const float* __restrict__ ssum, float* __restrict__ acc) {
    int lane = threadIdx.x & 31;
    int e    = (int)(((size_t)blockIdx.x * 256 + threadIdx.x) >> 5);
    if (e >= nE + N) return;
    int s, d; edge_nodes(ei, nE, e, s, d);
    int head    = lane >> 3;                   // (lane*4)/DHEAD
    float alpha = ebuf[e * HEADS + head] /
                  (ssum[d * HEADS + head] + 1e-16f);
    const float4 hv = ((const float4*)(hp + (size_t)s * HID))[lane];
    float* o = acc + (size_t)d * HID + lane * 4;
    atomicAdd(o + 0, hv.x * alpha);
    atomicAdd(o + 1, hv.y * alpha);
    atomicAdd(o + 2, hv.z * alpha);
    atomicAdd(o + 3, hv.w * alpha);
}

__global__ void bias_relu_kernel(const float* __restrict__ acc,
                                 const float* __restrict__ bias,
                                 float* __restrict__ out, int total) {
    int t = blockIdx.x * blockDim.x + threadIdx.x;
    if (t >= total) return;
    out[t] = fmaxf(acc[t] + bias[t & (HID - 1)], 0.f);
}

// ------------------------- global add pool ---------------------------------
__global__ void pool_kernel(const float* __restrict__ h,
                            const int* __restrict__ batch,
                            float* __restrict__ g, int N) {
    int t = blockIdx.x * blockDim.x + threadIdx.x;
    if (t >= N * HID) return;
    int n = t >> 7, j = t & (HID - 1);
    atomicAdd(&g[(size_t)batch[n] * HID + j], h[t]);
}

// ------------------------- classifier head ---------------------------------
__global__ __launch_bounds__(128) void head_kernel(
        const float* __restrict__ z, const float* __restrict__ scale,
        const float* __restrict__ shift, const float* __restrict__ clsw,
        const float* __restrict__ clsb, float* __restrict__ out) {
    int g = blockIdx.x * 128 + threadIdx.x;
    if (g >= NGRAPHS) return;
    float logit[NCLS];
#pragma unroll
    for (int c = 0; c < NCLS; ++c) logit[c] = clsb[c];
    for (int j = 0; j < HID; ++j) {
        float v = z[(size_t)g * HID + j] * scale[j] + shift[j];
#pragma unroll
        for (int c = 0; c < NCLS; ++c) logit[c] += v * clsw[j * NCLS + c];
    }
    float m = logit[0];
#pragma unroll
    for (int c = 1; c < NCLS; ++c) m = fmaxf(m, logit[c]);
    float s = 0.f;
#pragma unroll
    for (int c = 0; c < NCLS; ++c) s += expf(logit[c] - m);
    float lse = m + logf(s);
#pragma unroll
    for (int c = 0; c < NCLS; ++c) out[g * NCLS + c] = logit[c] - lse;
}

// ---------------------------------------------------------------------------
extern "C" void kernel_launch(void* const* d_in, const int* in_sizes, int n_in,
                              void* d_out, int out_size, void* d_ws, size_t ws_size,
                              hipStream_t stream) {
    const float* x       = (const float*)d_in[0];
    const int*   ei      = (const int*)  d_in[1];
    const int*   batch   = (const int*)  d_in[2];
    const float* w_feat  = (const float*)d_in[3];
    const float* bnf_g   = (const float*)d_in[4];
    const float* bnf_b   = (const float*)d_in[5];
    const float* bnc_g   = (const float*)d_in[6];   // [3,128]
    const float* bnc_b   = (const float*)d_in[7];
    const float* gat_w   = (const float*)d_in[8];   // [3,128,128]
    const float* gat_as  = (const float*)d_in[9];   // [3,4,32]
    const float* gat_ad  = (const float*)d_in[10];
    const float* gat_b   = (const float*)d_in[11];  // [3,128]
    const float* bnfc_g  = (const float*)d_in[12];
    const float* bnfc_b  = (const float*)d_in[13];
    const float* lin_w   = (const float*)d_in[14];
    const float* lin_b   = (const float*)d_in[15];
    const float* bnh_g   = (const float*)d_in[16];
    const float* bnh_b   = (const float*)d_in[17];
    const float* cls_w   = (const float*)d_in[18];
    const float* cls_b   = (const float*)d_in[19];
    float* out = (float*)d_out;

    const int N  = in_sizes[0] / HID;   // 50000
    const int nE = in_sizes[1] / 2;     // 800000
    const int Ec = nE + N;              // with self-loops
    (void)n_in; (void)out_size; (void)ws_size;

    // ---- bump allocator over workspace ----
    char* ws = (char*)d_ws;
    size_t off = 0;
    auto alloc = [&](size_t bytes) -> void* {
        void* p = ws + off;
        off += (bytes + 255) & ~(size_t)255;
        return p;
    };
    float*    buf_h   = (float*)alloc((size_t)N * HID * 4);
    float*    buf_hp  = (float*)alloc((size_t)N * HID * 4);
    float*    buf_acc = (float*)alloc((size_t)N * HID * 4);
    float*    ebuf    = (float*)alloc((size_t)Ec * HEADS * 4);
    float*    als     = (float*)alloc((size_t)N * HEADS * 4);
    float*    ald     = (float*)alloc((size_t)N * HEADS * 4);
    unsigned* mkey    = (unsigned*)alloc((size_t)N * HEADS * 4);
    float*    ssum    = (float*)alloc((size_t)N * HEADS * 4);
    float*    sumb    = (float*)alloc(HID * 4);
    float*    sumsq   = (float*)alloc(HID * 4);
    float*    scale   = (float*)alloc(HID * 4);
    float*    shift   = (float*)alloc(HID * 4);
    float*    wfold   = (float*)alloc(HID * HID * 4);
    float*    bfold   = (float*)alloc(HID * 4);
    float*    gpool   = (float*)alloc((size_t)NGRAPHS * HID * 4);
    float*    g2      = (float*)alloc((size_t)NGRAPHS * HID * 4);
    float*    scale2  = (float*)alloc(HID * 4);
    float*    shift2  = (float*)alloc(HID * 4);

    auto fill = [&](void* p, unsigned v, int n) {
        fill_u32_kernel<<<(n + 255) / 256, 256, 0, stream>>>((unsigned*)p, v, n);
    };
    auto bn_fold = [&](const float* src, int rows, const float* g_, const float* b_,
                       const float* W_, const float* bias_in,
                       float* sc, float* sh) {
        fill(sumb, 0u, HID); fill(sumsq, 0u, HID);
        colsum_kernel<<<(rows + 127) / 128, 128, 0, stream>>>(src, rows, sumb, sumsq);
        bn_scale_shift_kernel<<<1, 128, 0, stream>>>(sumb, sumsq, g_, b_,
                                                     1.f / (float)rows, sc, sh);
        if (W_) fold_weight_kernel<<<1, 128, 0, stream>>>(W_, sc, sh, bias_in,
                                                          wfold, bfold);
    };

    const unsigned NEG_INF_KEY = ~0xFF800000u;   // orderable key of -inf

    // ---- stage 1: h = relu(bn(x) @ w_feat) ----
    bn_fold(x, N, bnf_g, bnf_b, w_feat, nullptr, scale, shift);
    gemm128_wmma_kernel<<<(N + 63) / 64, 128, 0, stream>>>(x, wfold, bfold,
                                                           buf_h, N, 1);

    // ---- stage 2: 3x GAT conv ----
    for (int i = 0; i < 3; ++i) {
        bn_fold(buf_h, N, bnc_g + i * HID, bnc_b + i * HID,
                gat_w + (size_t)i * HID * HID, nullptr, scale, shift);
        gemm128_wmma_kernel<<<(N + 63) / 64, 128, 0, stream>>>(buf_h, wfold, bfold,
                                                               buf_hp, N, 0);
        alpha_kernel<<<(N * HEADS + 255) / 256, 256, 0, stream>>>(
            buf_hp, gat_as + i * HEADS * DHEAD, gat_ad + i * HEADS * DHEAD,
            als, ald, N);

        fill(mkey, NEG_INF_KEY, N * HEADS);
        fill(ssum, 0u, N * HEADS);
        const int ethreads = Ec * HEADS;
        edge_max_kernel<<<(ethreads + 255) / 256, 256, 0, stream>>>(
            ei, nE, N, als, ald, mkey);
        edge_exp_kernel<<<(ethreads + 255) / 256, 256, 0, stream>>>(
            ei, nE, N, als, ald, mkey, ebuf, ssum);

        fill(buf_acc, 0u, N * HID);
        scatter_kernel<<<(int)(((size_t)Ec * 32 + 255) / 256), 256, 0, stream>>>(
            ei, nE, N, buf_hp, ebuf, ssum, buf_acc);
        bias_relu_kernel<<<(N * HID + 255) / 256, 256, 0, stream>>>(
            buf_acc, gat_b + i * HID, buf_h, N * HID);
    }

    // ---- stage 3: pool + MLP head ----
    fill(gpool, 0u, NGRAPHS * HID);
    pool_kernel<<<(N * HID + 255) / 256, 256, 0, stream>>>(buf_h, batch, gpool, N);

    bn_fold(gpool, NGRAPHS, bnfc_g, bnfc_b, lin_w, lin_b, scale, shift);
    gemm128_wmma_kernel<<<(NGRAPHS + 63) / 64, 128, 0, stream>>>(gpool, wfold,
                                                                 bfold, g2,
                                                                 NGRAPHS, 1);
    bn_fold(g2, NGRAPHS, bnh_g, bnh_b, nullptr, nullptr, scale2, shift2);
    head_kernel<<<2, 128, 0, stream>>>(g2, scale2, shift2, cls_w, cls_b, out);
}